// HeteroGNN_HGT_79448305041989
// MI455X (gfx1250) — compile-verified
//
#include <hip/hip_runtime.h>
#include <math.h>

typedef float v2f __attribute__((ext_vector_type(2)));
typedef float v8f __attribute__((ext_vector_type(8)));

#define NA 50000
#define NP 100000
#define NTOT 150000
#define HID 128

// ---- order-preserving float<->uint encoding for atomic max ----
static __device__ __forceinline__ unsigned encf(float f) {
    unsigned u = __float_as_uint(f);
    return (u & 0x80000000u) ? ~u : (u | 0x80000000u);
}
static __device__ __forceinline__ float decf(unsigned v) {
    return (v & 0x80000000u) ? __uint_as_float(v & 0x7fffffffu) : __uint_as_float(~v);
}
#define ENC_NEG_INF 0x007FFFFFu   // encf(-inf)

// ---- CDNA5 async global->LDS DMA (ASYNCcnt-tracked, no VGPR round trip) ----
// Flat pointers into the LDS aperture carry the LDS byte offset in their low
// 32 bits (ISA 10.2), so truncation yields the VDST LDS address.
static __device__ __forceinline__ void async_load_b128(const float* g, float* l) {
    unsigned loff = (unsigned)(size_t)(void*)l;
    asm volatile("global_load_async_to_lds_b128 %0, %1, off"
                 :: "v"(loff), "v"(g) : "memory");
}
static __device__ __forceinline__ void wait_async_all() {
    asm volatile("s_wait_asynccnt 0x0" ::: "memory");
}

// =====================================================================
// GEMM: Y[M,N](ldy) = act( X[M,K](lda) @ W[K,N] + bias )
// V_WMMA_F32_16X16X4_F32 (full fp32 matrix pipe).
// Block = 256 threads (8 waves), block tile = 64 rows x N cols.
// CT = N/16 column tiles; each wave owns one 16-col slab and
// RPW = CT/2 row tiles, so each B fragment feeds RPW WMMAs.
// X tile is staged in LDS via global_load_async_to_lds_b128.
// Requires K%4==0, K multiple of 64, K<=256.
// =====================================================================
template <int CT>
__global__ void gemm_wmma_f32_t(const float* __restrict__ X, int lda,
                                const float* __restrict__ W,
                                const float* __restrict__ bias,
                                float* __restrict__ Y, int ldy,
                                int M, int K, int act)
{
    constexpr int N   = CT * 16;
    constexpr int RPW = CT / 2;          // 4*CT tiles / 8 waves
    extern __shared__ float lds[];
    const int pitch = K + 4;             // K%64==0 -> 4-bank stagger, 16B-aligned rows
    const int row0  = blockIdx.x * 64;
    const int tid   = threadIdx.x;

    // ---- stage 64 x K tile of X into LDS with async DMA (zero-pad OOB rows)
    const int vecPerRow = K >> 2;                    // float4 per row
    const int totalVec  = 64 * vecPerRow;
    for (int i = tid; i < totalVec; i += 256) {
        int r  = i / vecPerRow;
        int c4 = i - r * vecPerRow;
        float* lp = &lds[r * pitch + c4 * 4];
        int grow = row0 + r;
        if (grow < M) {
            async_load_b128(X + (size_t)grow * lda + c4 * 4, lp);
        } else {
            *(float4*)lp = make_float4(0.f, 0.f, 0.f, 0.f);
        }
    }
    wait_async_all();
    __syncthreads();

    const int wave = tid >> 5, lane = tid & 31;
    const int half = lane >> 4, lr = lane & 15;
    const int colBase     = (wave % CT) * 16;
    const int rowTileBase = (wave / CT) * RPW;

    v8f acc[RPW];
#pragma unroll
    for (int rt = 0; rt < RPW; ++rt)
        acc[rt] = (v8f){0.f, 0.f, 0.f, 0.f, 0.f, 0.f, 0.f, 0.f};

#pragma unroll 2
    for (int kk = 0; kk < K; kk += 4) {
        // B frag (ISA: VGPR0 = B-row krow, VGPR1 = B-row krow+1, col = lane%16)
        int krow = kk + 2 * half;
        v2f b;
        b.x = W[(size_t)krow * N + colBase + lr];
        b.y = W[(size_t)(krow + 1) * N + colBase + lr];
#pragma unroll
        for (int rt = 0; rt < RPW; ++rt) {
            // A frag: lanes0-15 K={kk,kk+1}, lanes16-31 K={kk+2,kk+3} (8B aligned)
            const v2f a = *(const v2f*)&lds[((rowTileBase + rt) * 16 + lr) * pitch + krow];
            acc[rt] = __builtin_amdgcn_wmma_f32_16x16x4_f32(false, a, false, b,
                                                            (short)0, acc[rt], false, false);
        }
    }

    float bv = bias ? bias[colBase + lr] : 0.f;
#pragma unroll
    for (int rt = 0; rt < RPW; ++rt) {
        for (int rr = 0; rr < 8; ++rr) {
            int row = row0 + (rowTileBase + rt) * 16 + rr + 8 * half;  // D: VGPR r -> M=r / r+8
            if (row < M) {
                float v = acc[rt][rr] + bv;
                if (act == 1) v = fmaxf(v, 0.f);
                Y[(size_t)row * ldy + colBase + lr] = v;
            }
        }
    }
}

// =====================================================================
// Edge attention phase (per edge type). Q/KR rows fit in 192MB L2.
// =====================================================================
__global__ void edge_logits(const int* __restrict__ src, const int* __restrict__ dst,
                            const float* __restrict__ Q,   // dst-type q rows
                            const float* __restrict__ KR,  // src-type k_rel rows
                            const float* __restrict__ p_rel,
                            float* __restrict__ logit,     // [E*2]
                            unsigned* __restrict__ segmax, // [Ndst*2] encoded
                            int E)
{
    int gw   = blockIdx.x * (blockDim.x >> 5) + (threadIdx.x >> 5);
    int lane = threadIdx.x & 31;
    int e = gw >> 1, h = gw & 1;
    if (e >= E) return;
    int s = src[e], d = dst[e];
    const float2* qrow = (const float2*)(Q  + (size_t)d * HID + h * 64);
    const float2* krow = (const float2*)(KR + (size_t)s * HID + h * 64);
    float2 qv = qrow[lane];
    float2 kv = krow[lane];
    float p = qv.x * kv.x + qv.y * kv.y;
    for (int off = 16; off > 0; off >>= 1) p += __shfl_xor(p, off, 32);
    if (lane == 0) {
        float lg = p * p_rel[h] * 0.125f;     // /sqrt(64)
        logit[(size_t)e * 2 + h] = lg;
        atomicMax(&segmax[(size_t)d * 2 + h], encf(lg));
    }
}

__global__ void edge_expsum(const int* __restrict__ dst,
                            float* __restrict__ logit,
                            const unsigned* __restrict__ segmax,
                            float* __restrict__ segsum, int E)
{
    int i = blockIdx.x * blockDim.x + threadIdx.x;   // over E*2
    if (i >= E * 2) return;
    int e = i >> 1, h = i & 1;
    int d = dst[e];
    float m  = decf(segmax[(size_t)d * 2 + h]);
    float ev = expf(logit[i] - m);
    logit[i] = ev;                                   // overwrite with exp value
    atomicAdd(&segsum[(size_t)d * 2 + h], ev);
}

// block 256 = 2 edges x 128 channels; scatter-add messages
__global__ void edge_message(const int* __restrict__ src, const int* __restrict__ dst,
                             const float* __restrict__ VR,
                             const float* __restrict__ logit,
                             const float* __restrict__ segsum,
                             float* __restrict__ out, int E)
{
    int e = blockIdx.x * 2 + (threadIdx.x >> 7);
    int c = threadIdx.x & 127;
    if (e >= E) return;
    int s = src[e], d = dst[e];
    int h = c >> 6;
    float a = logit[(size_t)e * 2 + h] / (segsum[(size_t)d * 2 + h] + 1e-16f);
    atomicAdd(&out[(size_t)d * HID + c], VR[(size_t)s * HID + c] * a);
}

// =====================================================================
// Elementwise / pooling / tail kernels
// =====================================================================
__global__ void fill_f32(float* __restrict__ p, float v, size_t n) {
    size_t i = (size_t)blockIdx.x * blockDim.x + threadIdx.x;
    if (i < n) p[i] = v;
}
__global__ void fill_u32(unsigned* __restrict__ p, unsigned v, size_t n) {
    size_t i = (size_t)blockIdx.x * blockDim.x + threadIdx.x;
    if (i < n) p[i] = v;
}
__global__ void gelu_inplace(float* __restrict__ p, size_t n) {
    size_t i = (size_t)blockIdx.x * blockDim.x + threadIdx.x;
    if (i < n) {
        float x = p[i];
        p[i] = 0.5f * x * (1.0f + erff(x * 0.70710678118654752f));
    }
}
__global__ void blend_skip(float* __restrict__ y, const float* __restrict__ xold,
                           const float* __restrict__ skip, size_t n) {
    size_t i = (size_t)blockIdx.x * blockDim.x + threadIdx.x;
    if (i < n) {
        float beta = 1.0f / (1.0f + expf(-skip[0]));
        y[i] = beta * y[i] + (1.0f - beta) * xold[i];
    }
}
__global__ void pool_sum(const float* __restrict__ X, const int* __restrict__ batch,
                         float* __restrict__ pooled, int colOff, size_t n) {
    size_t i = (size_t)blockIdx.x * blockDim.x + threadIdx.x;
    if (i < n) {
        int node = (int)(i >> 7), c = (int)(i & 127);
        atomicAdd(&pooled[(size_t)batch[node] * 256 + colOff + c], X[i]);
    }
}
__global__ void final_lin(const float* __restrict__ Hm, const float* __restrict__ W,
                          const float* __restrict__ b, float* __restrict__ out) {
    int g = threadIdx.x;   // 64 graphs
    if (g < 64) {
        float s = b[0];
        for (int c = 0; c < 128; ++c) s += Hm[(size_t)g * 128 + c] * W[c];
        out[g] = s;
    }
}

// =====================================================================
// Host orchestration
// =====================================================================
extern "C" void kernel_launch(void* const* d_in, const int* in_sizes, int n_in,
                              void* d_out, int out_size, void* d_ws, size_t ws_size,
                              hipStream_t stream) {
    (void)in_sizes; (void)n_in; (void)out_size; (void)ws_size;
    auto F = [&](int i) { return (const float*)d_in[i]; };
    auto I = [&](int i) { return (const int*)d_in[i]; };

    // ---- workspace layout (floats) ----
    float* ws = (float*)d_ws;
    size_t o = 0;
    float* X0   = ws + o; o += (size_t)NTOT * HID;
    float* X1   = ws + o; o += (size_t)NTOT * HID;
    float* KB   = ws + o; o += (size_t)NTOT * HID;
    float* QB   = ws + o; o += (size_t)NTOT * HID;
    float* VB   = ws + o; o += (size_t)NTOT * HID;
    float* OUTB = ws + o; o += (size_t)NTOT * HID;
    float* KR   = ws + o; o += (size_t)NP * HID;
    float* VR   = ws + o; o += (size_t)NP * HID;
    float* LOG  = ws + o; o += (size_t)300000 * 2;
    unsigned* SEGMAX = (unsigned*)(ws + o); o += (size_t)NP * 2;
    float* SEGSUM = ws + o; o += (size_t)NP * 2;
    float* POOL  = ws + o; o += (size_t)64 * 256;
    float* HBUF  = ws + o; o += (size_t)64 * 128;

    auto gemm = [&](const float* X, int lda, const float* W, const float* b,
                    float* Y, int ldy, int M, int K, int N, int act) {
        size_t shmem = (size_t)64 * (K + 4) * sizeof(float);
        dim3 g((M + 63) / 64);
        if (N == 128)
            gemm_wmma_f32_t<8><<<g, dim3(256), shmem, stream>>>(X, lda, W, b, Y, ldy, M, K, act);
        else
            gemm_wmma_f32_t<4><<<g, dim3(256), shmem, stream>>>(X, lda, W, b, Y, ldy, M, K, act);
    };
    auto fillf = [&](float* p, float v, size_t n) {
        fill_f32<<<dim3((unsigned)((n + 255) / 256)), dim3(256), 0, stream>>>(p, v, n);
    };
    auto fillu = [&](unsigned* p, unsigned v, size_t n) {
        fill_u32<<<dim3((unsigned)((n + 255) / 256)), dim3(256), 0, stream>>>(p, v, n);
    };

    // ---- input projections + ReLU (63/64 lin_in.author W/b ; 65/66 lin_in.paper W/b)
    gemm(F(0), 64,  F(63), F(64), X0,                     HID, NA, 64,  HID, 1);
    gemm(F(1), 128, F(65), F(66), X0 + (size_t)NA * HID,  HID, NP, 128, HID, 1);

    // edge-type tables
    const int*   eiv[3]   = { I(2), I(3), I(4) };
    const int    Ee[3]    = { 150000, 300000, 150000 };
    const int    srcOff[3]= { 0, NA, NA };
    const int    NsrcA[3] = { NA, NP, NP };
    const int    dstOff[3]= { NA, NA, 0 };
    const int    NdstA[3] = { NP, NP, NA };

    float* xcur = X0;
    float* xnxt = X1;

    for (int l = 0; l < 2; ++l) {
        int Lb = 7 + l * 27;
        const int kW[2] = { Lb + 7,  Lb + 9  }, kb[2] = { Lb + 8,  Lb + 10 };
        const int qW[2] = { Lb + 17, Lb + 19 }, qb[2] = { Lb + 18, Lb + 20 };
        const int vW[2] = { Lb + 23, Lb + 25 }, vb[2] = { Lb + 24, Lb + 26 };
        const int aW[2] = { Lb + 0,  Lb + 2  }, ab[2] = { Lb + 1,  Lb + 3  };
        const int skipI[2] = { Lb + 21, Lb + 22 };
        const size_t rowOff[2] = { 0, (size_t)NA * HID };
        const int    Nn[2]     = { NA, NP };

        for (int t = 0; t < 2; ++t) {
            gemm(xcur + rowOff[t], HID, F(kW[t]), F(kb[t]), KB + rowOff[t], HID, Nn[t], HID, HID, 0);
            gemm(xcur + rowOff[t], HID, F(qW[t]), F(qb[t]), QB + rowOff[t], HID, Nn[t], HID, HID, 0);
            gemm(xcur + rowOff[t], HID, F(vW[t]), F(vb[t]), VB + rowOff[t], HID, Nn[t], HID, HID, 0);
        }

        fillf(OUTB, 0.f, (size_t)NTOT * HID);

        for (int t = 0; t < 3; ++t) {
            const float* a_rel = F(Lb + 4 + t);   // [2,64,64]
            const float* m_rel = F(Lb + 11 + t);
            const float* p_rel = F(Lb + 14 + t);  // [2]
            int E = Ee[t], Nsrc = NsrcA[t], Ndst = NdstA[t];
            const float* Kb = KB + (size_t)srcOff[t] * HID;
            const float* Vb = VB + (size_t)srcOff[t] * HID;
            const float* Qb = QB + (size_t)dstOff[t] * HID;
            float* Ob = OUTB + (size_t)dstOff[t] * HID;
            const int* srcp = eiv[t];
            const int* dstp = eiv[t] + E;

            // per-head relation transforms (64x64 GEMMs on source nodes)
            for (int h = 0; h < 2; ++h) {
                gemm(Kb + h * 64, HID, a_rel + (size_t)h * 64 * 64, nullptr,
                     KR + h * 64, HID, Nsrc, 64, 64, 0);
                gemm(Vb + h * 64, HID, m_rel + (size_t)h * 64 * 64, nullptr,
                     VR + h * 64, HID, Nsrc, 64, 64, 0);
            }

            fillu(SEGMAX, ENC_NEG_INF, (size_t)Ndst * 2);
            fillf(SEGSUM, 0.f, (size_t)Ndst * 2);

            int waves = E * 2;
            edge_logits<<<dim3((waves + 7) / 8), dim3(256), 0, stream>>>(
                srcp, dstp, Qb, KR, p_rel, LOG, SEGMAX, E);
            edge_expsum<<<dim3((E * 2 + 255) / 256), dim3(256), 0, stream>>>(
                dstp, LOG, SEGMAX, SEGSUM, E);
            edge_message<<<dim3((E + 1) / 2), dim3(256), 0, stream>>>(
                srcp, dstp, VR, LOG, SEGSUM, Ob, E);
        }

        // newx = sigmoid(skip)*(gelu(out) @ Wa + ba) + (1-sigmoid(skip))*x
        gelu_inplace<<<dim3((unsigned)(((size_t)NTOT * HID + 255) / 256)), dim3(256), 0, stream>>>(
            OUTB, (size_t)NTOT * HID);
        for (int t = 0; t < 2; ++t) {
            gemm(OUTB + rowOff[t], HID, F(aW[t]), F(ab[t]), xnxt + rowOff[t], HID, Nn[t], HID, HID, 0);
            size_t n = (size_t)Nn[t] * HID;
            blend_skip<<<dim3((unsigned)((n + 255) / 256)), dim3(256), 0, stream>>>(
                xnxt + rowOff[t], xcur + rowOff[t], F(skipI[t]), n);
        }
        float* tmp = xcur; xcur = xnxt; xnxt = tmp;
    }

    // ---- per-graph sum pooling (author -> cols 0..127, paper -> 128..255)
    fillf(POOL, 0.f, (size_t)64 * 256);
    pool_sum<<<dim3((unsigned)(((size_t)NA * HID + 255) / 256)), dim3(256), 0, stream>>>(
        xcur, I(5), POOL, 0, (size_t)NA * HID);
    pool_sum<<<dim3((unsigned)(((size_t)NP * HID + 255) / 256)), dim3(256), 0, stream>>>(
        xcur + (size_t)NA * HID, I(6), POOL, 128, (size_t)NP * HID);

    // ---- mlp: [64,256] @ [256,128] + b, then lin: [64,128] @ [128,1] + b
    gemm(POOL, 256, F(67), F(68), HBUF, 128, 64, 256, 128, 0);
    final_lin<<<dim3(1), dim3(64), 0, stream>>>(HBUF, F(61), F(62), (float*)d_out);
}